// IMPALAFruitfly_54795192762759
// MI455X (gfx1250) — compile-verified
//
#include <hip/hip_runtime.h>
#include <hip/hip_bf16.h>
#include <math.h>

#define VOCAB 30522
#define TWOV  61044
#define KC    2048
#define TOPK  32
#define NOUT  18
#define BATCH 256
#define SEQL  128

typedef __attribute__((ext_vector_type(2))) float v2f;
typedef __attribute__((ext_vector_type(8))) float v8f;

// ---------------------------------------------------------------------------
// Kernel 1: acts[b,k] = sum_l (ids[b,l] != 0) * (W[k, id] + W[k, id+VOCAB])
// Grid: (BATCH fast, KC/256 slow) so concurrent blocks share one k-tile's
// 62MB W footprint in the 192MB L2; each W cacheline hits HBM ~once (~500MB).
// ---------------------------------------------------------------------------
__global__ __launch_bounds__(256) void ff_acts_kernel(const int* __restrict__ ids,
                                                      const float* __restrict__ W,
                                                      float* __restrict__ acts)
{
    __shared__ int s_ids[SEQL];
    const int b   = blockIdx.x;
    const int kt  = blockIdx.y;
    const int tid = threadIdx.x;

    if (tid < SEQL) s_ids[tid] = ids[b * SEQL + tid];
    __syncthreads();

    const int k = kt * 256 + tid;
    const float* __restrict__ Wk = W + (size_t)k * (size_t)TWOV;

    float a0 = 0.f, a1 = 0.f, a2 = 0.f, a3 = 0.f;
    for (int l = 0; l < SEQL; l += 4) {
        const int i0 = s_ids[l + 0];
        const int i1 = s_ids[l + 1];
        const int i2 = s_ids[l + 2];
        const int i3 = s_ids[l + 3];
        const float m0 = (i0 != 0) ? 1.0f : 0.0f;
        const float m1 = (i1 != 0) ? 1.0f : 0.0f;
        const float m2 = (i2 != 0) ? 1.0f : 0.0f;
        const float m3 = (i3 != 0) ? 1.0f : 0.0f;
        a0 = fmaf(m0, Wk[i0] + Wk[i0 + VOCAB], a0);
        a1 = fmaf(m1, Wk[i1] + Wk[i1 + VOCAB], a1);
        a2 = fmaf(m2, Wk[i2] + Wk[i2 + VOCAB], a2);
        a3 = fmaf(m3, Wk[i3] + Wk[i3 + VOCAB], a3);
    }
    acts[(size_t)b * KC + k] = (a0 + a1) + (a2 + a3);
}

// ---------------------------------------------------------------------------
// Kernel 2: per-row top-32 -> binary hash (0/1 f32), lowest-index tie-break
// like jax.lax.top_k. One 256-thread block per row; acts row staged in LDS.
// ---------------------------------------------------------------------------
__global__ __launch_bounds__(256) void ff_topk_kernel(const float* __restrict__ acts,
                                                      float* __restrict__ bh)
{
    __shared__ float vals[KC];
    __shared__ unsigned long long red[256];
    const int row = blockIdx.x;
    const int tid = threadIdx.x;

    #pragma unroll
    for (int j = 0; j < KC / 256; ++j) {
        const int idx = tid + 256 * j;
        vals[idx] = acts[(size_t)row * KC + idx];
        bh[(size_t)row * KC + idx] = 0.0f;   // fully re-init every call
    }
    __syncthreads();

    for (int t = 0; t < TOPK; ++t) {
        // local max over 8 strided elements, key = (monotone f32) << 32 | (2047-idx)
        unsigned long long best = 0ull;
        #pragma unroll
        for (int j = 0; j < KC / 256; ++j) {
            const int idx = tid + 256 * j;
            unsigned u = __float_as_uint(vals[idx]);
            u = (u & 0x80000000u) ? ~u : (u | 0x80000000u);   // order-preserving
            const unsigned long long key =
                ((unsigned long long)u << 32) | (unsigned)(KC - 1 - idx);
            if (key > best) best = key;
        }
        red[tid] = best;
        __syncthreads();
        #pragma unroll
        for (int s = 128; s > 0; s >>= 1) {
            if (tid < s) {
                const unsigned long long o = red[tid + s];
                if (o > red[tid]) red[tid] = o;
            }
            __syncthreads();
        }
        const int widx = (KC - 1) - (int)(red[0] & 0xFFFFFFFFull);
        if (tid == 0) {
            vals[widx] = -INFINITY;                     // exclude from next rounds
            bh[(size_t)row * KC + widx] = 1.0f;
        }
        __syncthreads();
    }
}

// ---------------------------------------------------------------------------
// Kernel 3: logits[256,18] = bh[256,2048] @ Wa^T[2048,18] + ba, via
// V_WMMA_F32_16X16X4_F32. One wave per 16x16 output tile; N padded 18->32
// with clamped/masked B loads (no EXEC divergence at the WMMA).
//  A 16x4 layout: lane%16 = M, half = lane/16 carries K{0,1} vs K{2,3}.
//  B 4x16 / D 16x16: rows striped across lanes within a VGPR.
// ---------------------------------------------------------------------------
__global__ __launch_bounds__(32) void ff_logits_wmma_kernel(const float* __restrict__ bh,
                                                            const float* __restrict__ Wa,
                                                            const float* __restrict__ ba,
                                                            float* __restrict__ out)
{
    const int lane = threadIdx.x;
    const int half = lane >> 4;
    const int l16  = lane & 15;

    const int mrow = blockIdx.x * 16 + l16;          // A row this lane feeds
    const int n    = blockIdx.y * 16 + l16;          // B column this lane feeds
    const float mB = (n < NOUT) ? 1.0f : 0.0f;
    const int  nc  = (n < NOUT) ? n : (NOUT - 1);    // clamp: no OOB reads

    const float* __restrict__ Arow = bh + (size_t)mrow * KC;
    const float* __restrict__ Brow = Wa + (size_t)nc * KC;

    v8f acc = {};
    for (int k = 0; k < KC; k += 4) {
        const int kk = k + half * 2;
        v2f a  = *(const v2f*)(Arow + kk);           // A[M][kk], A[M][kk+1]
        v2f bb = *(const v2f*)(Brow + kk);           // Wa[n][kk] = B[kk][n]
        bb *= mB;                                    // zero out padded columns
        acc = __builtin_amdgcn_wmma_f32_16x16x4_f32(
            /*neg_a=*/false, a, /*neg_b=*/false, bb,
            /*c_mod=*/(short)0, acc, /*reuse_a=*/false, /*reuse_b=*/false);
    }

    #pragma unroll
    for (int j = 0; j < 8; ++j) {
        const int M = blockIdx.x * 16 + j + half * 8;
        const int N = blockIdx.y * 16 + l16;
        if (N < NOUT) out[M * NOUT + N] = acc[j] + ba[N];
    }
}

// ---------------------------------------------------------------------------
extern "C" void kernel_launch(void* const* d_in, const int* in_sizes, int n_in,
                              void* d_out, int out_size, void* d_ws, size_t ws_size,
                              hipStream_t stream) {
    (void)in_sizes; (void)n_in; (void)out_size; (void)ws_size;

    const int*   ids = (const int*)  d_in[0];   // [256,128] int32
    const float* W   = (const float*)d_in[1];   // [2048, 61044] f32
    const float* Wa  = (const float*)d_in[2];   // [18, 2048] f32
    const float* ba  = (const float*)d_in[3];   // [18] f32
    float*       out = (float*)d_out;           // [256, 18] f32

    float* acts = (float*)d_ws;                       // 256*2048 f32 = 2MB
    float* bh   = acts + (size_t)BATCH * KC;          // 256*2048 f32 = 2MB

    // 1) gather-reduce (memory bound, ~500MB compulsory HBM -> ~21us)
    ff_acts_kernel<<<dim3(BATCH, KC / 256), 256, 0, stream>>>(ids, W, acts);
    // 2) per-row top-32 -> binary hash
    ff_topk_kernel<<<dim3(BATCH), 256, 0, stream>>>(acts, bh);
    // 3) tiny f32 WMMA GEMM for the 18 logits
    ff_logits_wmma_kernel<<<dim3(BATCH / 16, 2), 32, 0, stream>>>(bh, Wa, ba, out);
}